// PPRPowerIteration_13511967113721
// MI455X (gfx1250) — compile-verified
//
#include <hip/hip_runtime.h>

#define N_NODES 50000
#define D_FEAT  128
#define ALPHA_F 0.1f
#define NITER   10
#define WPB     8   // waves (rows) per 256-thread block

// ---------------------------------------------------------------------------
// CSR construction helpers (run once per call; graph-capture safe)
// ---------------------------------------------------------------------------
__global__ void zero_i32(int* __restrict__ p, int n) {
    int i = blockIdx.x * blockDim.x + threadIdx.x;
    if (i < n) p[i] = 0;
}

__global__ void hist_rows(const int* __restrict__ rows, int* __restrict__ counts, int nnz) {
    int e = blockIdx.x * blockDim.x + threadIdx.x;
    if (e < nnz) atomicAdd(&counts[rows[e]], 1);
}

// Single-block exclusive scan over counts (in place -> becomes cursor) and
// row_ptr. 50k elements in 49 chunks of 1024: microseconds, no extra kernels.
__global__ void __launch_bounds__(1024)
scan_counts(int* cnt_cursor, int* __restrict__ row_ptr, int n) {
    __shared__ int sh[1024];
    const int tid = threadIdx.x;
    int carry = 0;
    for (int base = 0; base < n; base += 1024) {
        int i = base + tid;
        int v = (i < n) ? cnt_cursor[i] : 0;
        sh[tid] = v;
        __syncthreads();
        // Hillis-Steele inclusive scan, 10 steps
        for (int off = 1; off < 1024; off <<= 1) {
            int y = (tid >= off) ? sh[tid - off] : 0;
            __syncthreads();
            sh[tid] += y;
            __syncthreads();
        }
        if (i < n) {
            int excl = carry + sh[tid] - v;
            row_ptr[i]    = excl;
            cnt_cursor[i] = excl;   // becomes the scatter cursor
        }
        carry += sh[1023];
        __syncthreads();
    }
    if (tid == 0) row_ptr[n] = carry;
}

// Pack (col, val) into 8B records at CSR positions.
__global__ void scatter_edges(const int* __restrict__ rows, const int* __restrict__ cols,
                              const float* __restrict__ vals, int* __restrict__ cursor,
                              int2* __restrict__ epack, int nnz) {
    int e = blockIdx.x * blockDim.x + threadIdx.x;
    if (e < nnz) {
        int p = atomicAdd(&cursor[rows[e]], 1);
        epack[p] = make_int2(cols[e], __float_as_int(vals[e]));
    }
}

// ---------------------------------------------------------------------------
// Hot loop: wave-per-row CSR SpMM,  Pout[r] = alpha*E[r] + sum val*Pin[col]
// Edge metadata staged via gfx1250 async global->LDS DMA (ASYNCcnt path).
// Inner loop unrolled 4x so each wave keeps 4 x 512B row-gathers in flight.
// ---------------------------------------------------------------------------
__global__ void __launch_bounds__(256)
ppr_spmm(const float4* __restrict__ Pin, const float4* __restrict__ E4,
         const int2* __restrict__ epack, const int* __restrict__ row_ptr,
         float4* __restrict__ Pout, int n) {
    __shared__ int2 smeta[WPB][32];   // 8B * 32 edges per wave

    const int lane = threadIdx.x & 31;
    const int w    = threadIdx.x >> 5;
    const int r    = blockIdx.x * WPB + w;
    if (r >= n) return;

    const int start = row_ptr[r];
    const int end   = row_ptr[r + 1];

    float4 acc = E4[r * 32 + lane];          // 32 lanes x float4 = one 512B row
    acc.x *= ALPHA_F; acc.y *= ALPHA_F; acc.z *= ALPHA_F; acc.w *= ALPHA_F;

    const unsigned lds_slot =
        (unsigned)(unsigned long long)(&smeta[w][lane]);   // low 32b = LDS offset

    for (int j = start; j < end; j += 32) {
        const int cnt = min(32, end - j);
        if (lane < cnt) {
            unsigned long long gaddr =
                (unsigned long long)(const void*)(&epack[j + lane]);
            // per-lane 8B async copy: 32 lanes -> 256B of edge meta into LDS
            asm volatile("global_load_async_to_lds_b64 %0, %1, off"
                         :: "v"(lds_slot), "v"(gaddr) : "memory");
        }
        asm volatile("s_wait_asynccnt 0" ::: "memory");

        int k = 0;
        // 4-way unroll: 4 independent 512B gathers in flight per wave
        for (; k + 4 <= cnt; k += 4) {
            int2 m0 = smeta[w][k + 0];       // uniform addr -> ds_load_b64 bcast
            int2 m1 = smeta[w][k + 1];
            int2 m2 = smeta[w][k + 2];
            int2 m3 = smeta[w][k + 3];
            float4 p0 = Pin[(long long)m0.x * 32 + lane];
            float4 p1 = Pin[(long long)m1.x * 32 + lane];
            float4 p2 = Pin[(long long)m2.x * 32 + lane];
            float4 p3 = Pin[(long long)m3.x * 32 + lane];
            float v0 = __int_as_float(m0.y);
            float v1 = __int_as_float(m1.y);
            float v2 = __int_as_float(m2.y);
            float v3 = __int_as_float(m3.y);
            acc.x += v0 * p0.x; acc.y += v0 * p0.y; acc.z += v0 * p0.z; acc.w += v0 * p0.w;
            acc.x += v1 * p1.x; acc.y += v1 * p1.y; acc.z += v1 * p1.z; acc.w += v1 * p1.w;
            acc.x += v2 * p2.x; acc.y += v2 * p2.y; acc.z += v2 * p2.z; acc.w += v2 * p2.w;
            acc.x += v3 * p3.x; acc.y += v3 * p3.y; acc.z += v3 * p3.z; acc.w += v3 * p3.w;
        }
        for (; k < cnt; ++k) {
            int2 m = smeta[w][k];
            float val = __int_as_float(m.y);
            float4 p  = Pin[(long long)m.x * 32 + lane];
            acc.x += val * p.x;
            acc.y += val * p.y;
            acc.z += val * p.z;
            acc.w += val * p.w;
        }
    }
    Pout[r * 32 + lane] = acc;
}

// ---------------------------------------------------------------------------
extern "C" void kernel_launch(void* const* d_in, const int* in_sizes, int n_in,
                              void* d_out, int out_size, void* d_ws, size_t ws_size,
                              hipStream_t stream) {
    const float* E    = (const float*)d_in[0];
    const int*   rows = (const int*)  d_in[1];
    const int*   cols = (const int*)  d_in[2];
    const float* vals = (const float*)d_in[3];
    const int n   = N_NODES;
    const int nnz = in_sizes[1];

    // Workspace layout (~33 MB): ping buffer, row_ptr, cursor/counts, epack
    char*  ws  = (char*)d_ws;
    size_t off = 0;
    float* Pws = (float*)(ws + off); off += (size_t)n * D_FEAT * sizeof(float);
    off = (off + 255) & ~(size_t)255;
    int* row_ptr = (int*)(ws + off); off += (size_t)(n + 1) * sizeof(int);
    off = (off + 255) & ~(size_t)255;
    int* cursor  = (int*)(ws + off); off += (size_t)n * sizeof(int);
    off = (off + 255) & ~(size_t)255;
    int2* epack  = (int2*)(ws + off);

    const int tpb = 256;

    // Build CSR once per call
    zero_i32     <<<(n   + tpb - 1) / tpb, tpb, 0, stream>>>(cursor, n);
    hist_rows    <<<(nnz + tpb - 1) / tpb, tpb, 0, stream>>>(rows, cursor, nnz);
    scan_counts  <<<1, 1024, 0, stream>>>(cursor, row_ptr, n);
    scatter_edges<<<(nnz + tpb - 1) / tpb, tpb, 0, stream>>>(rows, cols, vals,
                                                             cursor, epack, nnz);

    // 10 power iterations, ping-pong ws <-> d_out (last iter lands in d_out)
    float* Pfinal    = (float*)d_out;
    const float* cur = E;
    for (int t = 0; t < NITER; ++t) {
        float* nxt = (t & 1) ? Pfinal : Pws;
        ppr_spmm<<<(n + WPB - 1) / WPB, tpb, 0, stream>>>(
            (const float4*)cur, (const float4*)E, epack, row_ptr, (float4*)nxt, n);
        cur = nxt;
    }
}